// ComplexGraph_13271448944742
// MI455X (gfx1250) — compile-verified
//
#include <hip/hip_runtime.h>
#include <hip/hip_bf16.h>

#define N_NODES 3072
#define EMBED   256
#define KNEI    10
#define BIGF    1e10f
#define TILE    16                  /* node tile: 16 i-nodes x 16 j-nodes */
#define NTI     (N_NODES / TILE)    /* 192 node tiles per dimension */
#define JT      4                   /* j-tiles per wave (A/key reuse) */

typedef __attribute__((ext_vector_type(2))) float v2f;
typedef __attribute__((ext_vector_type(8))) float v8f;

// ---------------------------------------------------------------------------
// Kernel 1: node_attr = W_res[S] + sinusoidal positional embedding
// ---------------------------------------------------------------------------
__global__ void __launch_bounds__(256)
node_attr_kernel(const float* __restrict__ W_res,
                 const int*   __restrict__ S,
                 const int*   __restrict__ RP,
                 float* __restrict__ out) {
    int idx = blockIdx.x * blockDim.x + threadIdx.x;   // N*256 threads exactly
    int i = idx >> 8;
    int d = idx & 255;
    int f = d >> 1;
    // inv_freq = 10000^(-f/128) = exp(-f * ln(10000)/128)
    float invf = expf(-(float)f * (9.210340371976184f / 128.0f));
    float ang  = (float)RP[i] * invf;
    float v    = (d & 1) ? cosf(ang) : sinf(ang);
    out[idx] = W_res[S[i] * EMBED + d] + v;
}

// ---------------------------------------------------------------------------
// Kernel 2: masked min-channel-pair distance matrix D[N][N] via WMMA.
//   For a 16x16 node tile, run 16 x V_WMMA_F32_16X16X4_F32, one per channel
//   pair (c1,c2):  A(c1)[m] = (-2x,-2y,-2z, ||p||^2) of node (i0+m) channel c1,
//   B(c2)[n] = (x,y,z,1) of node (j0+n) channel c2, so
//     C[m][n] = ||p||^2 - 2 p.q .
//   min over c1: elementwise vector-min of accumulators (in-lane).
//   min over c2: add lane-local ||q(c2)||^2, then vector-min (in-lane).
//   => no cross-lane shuffles at all; every lane writes 8 coalesced outputs.
// ---------------------------------------------------------------------------
__global__ void __launch_bounds__(256)
dist_kernel(const float* __restrict__ X,     // [N,4,3]
            const int*   __restrict__ Seg,
            const int*   __restrict__ bid,
            float* __restrict__ D) {
    const int lane = threadIdx.x & 31;
    const int wave = threadIdx.x >> 5;
    const int w    = blockIdx.x * 8 + wave;          // global work id
    const int ti   = w / (NTI / JT);                 // i node-tile
    const int tj0  = (w - ti * (NTI / JT)) * JT;     // first j node-tile
    const int half = lane >> 4;                      // 0: K=0,1   1: K=2,3
    const int r    = lane & 15;
    const int i0   = ti * TILE;

    // ---- A fragments, one per i-channel c1 (loaded once, reused JT times)
    v2f A[4];
#pragma unroll
    for (int c1 = 0; c1 < 4; ++c1) {
        const float* Pa = X + ((i0 + r) * 4 + c1) * 3;
        float ax = Pa[0], ay = Pa[1], az = Pa[2];
        float rn = ax * ax + ay * ay + az * az;
        A[c1][0] = half ? (-2.0f * az) : (-2.0f * ax);
        A[c1][1] = half ? rn           : (-2.0f * ay);
    }

    // ---- mask keys for this lane's 8 output rows m = v + half*8
    int rb[8], rs[8];
#pragma unroll
    for (int v = 0; v < 8; ++v) {
        int ig = i0 + v + half * 8;
        rb[v] = bid[ig];
        rs[v] = Seg[ig];
    }

    for (int jt = 0; jt < JT; ++jt) {
        const int tj = tj0 + jt;
        const int j0 = tj * TILE;
        const int jg = j0 + r;                        // this lane's column node
        const int bj = bid[jg], sj = Seg[jg];

        v8f res;
#pragma unroll
        for (int v = 0; v < 8; ++v) res[v] = 3.0e38f;

#pragma unroll
        for (int c2 = 0; c2 < 4; ++c2) {
            const float* Pb = X + (jg * 4 + c2) * 3;
            float bx = Pb[0], by = Pb[1], bz = Pb[2];
            float cn = bx * bx + by * by + bz * bz;   // ||q(c2)||^2, lane-local
            v2f B;
            B[0] = half ? bz   : bx;
            B[1] = half ? 1.0f : by;

            v8f mc = {};
            mc = __builtin_amdgcn_wmma_f32_16x16x4_f32(
                    false, A[0], false, B, (short)0, mc, false, false);
#pragma unroll
            for (int c1 = 1; c1 < 4; ++c1) {
                v8f t = {};
                t = __builtin_amdgcn_wmma_f32_16x16x4_f32(
                        false, A[c1], false, B, (short)0, t, false, false);
#pragma unroll
                for (int v = 0; v < 8; ++v) mc[v] = fminf(mc[v], t[v]);
            }
#pragma unroll
            for (int v = 0; v < 8; ++v) res[v] = fminf(res[v], mc[v] + cn);
        }

        // ---- mask + sqrt + store: lane writes rows m=v+half*8, column jg
#pragma unroll
        for (int v = 0; v < 8; ++v) {
            int ig = i0 + v + half * 8;
            bool valid = (rb[v] == bj) && (rs[v] == sj);
            float dn = sqrtf(fmaxf(res[v], 0.0f));
            D[(long)ig * N_NODES + jg] =
                valid ? ((ig == jg) ? dn + BIGF : dn) : BIGF;
        }
    }
}

// ---------------------------------------------------------------------------
// Kernel 3: per-row top-10 smallest (branchless unrolled insertion, reg-only),
//   ties broken toward lower index (matches lax.top_k ordering).
//   Rows are L2-resident (37.7 MB D << 192 MB L2); scan via float4.
// ---------------------------------------------------------------------------
__global__ void __launch_bounds__(256)
topk_kernel(const float* __restrict__ D,
            float* __restrict__ out_dst,
            float* __restrict__ out_dist,
            float* __restrict__ out_valid) {
    int row = blockIdx.x * blockDim.x + threadIdx.x;
    if (row >= N_NODES) return;

    float best[KNEI];
    int   idxs[KNEI];
#pragma unroll
    for (int k = 0; k < KNEI; ++k) { best[k] = 3.4e38f; idxs[k] = 0; }

    const float4* drow = (const float4*)(D + (long)row * N_NODES);
    for (int j4 = 0; j4 < N_NODES / 4; ++j4) {
        float4 q = drow[j4];
        float e[4] = {q.x, q.y, q.z, q.w};
#pragma unroll
        for (int t = 0; t < 4; ++t) {
            float d = e[t];
            int   j = j4 * 4 + t;
            if (d < best[KNEI - 1]) {
#pragma unroll
                for (int k = KNEI - 1; k >= 1; --k) {
                    bool up   = best[k - 1] > d;          // insertion point above
                    bool here = !up && (best[k] > d);     // insert exactly here
                    float nv  = up ? best[k - 1] : (here ? d : best[k]);
                    int   ni  = up ? idxs[k - 1] : (here ? j : idxs[k]);
                    best[k] = nv; idxs[k] = ni;
                }
                if (best[0] > d) { best[0] = d; idxs[0] = j; }
            }
        }
    }

#pragma unroll
    for (int k = 0; k < KNEI; ++k) {
        float dv = best[k];
        out_dst  [row * KNEI + k] = (float)idxs[k];
        out_dist [row * KNEI + k] = dv;
        out_valid[row * KNEI + k] = (dv < BIGF) ? 1.0f : 0.0f;
    }
}

// ---------------------------------------------------------------------------
extern "C" void kernel_launch(void* const* d_in, const int* in_sizes, int n_in,
                              void* d_out, int out_size, void* d_ws, size_t ws_size,
                              hipStream_t stream) {
    const float* X     = (const float*)d_in[0];   // [N,4,3]
    const int*   S     = (const int*)  d_in[1];   // [N]
    const int*   RP    = (const int*)  d_in[2];   // [N]
    const int*   Seg   = (const int*)  d_in[3];   // [N]
    const int*   bid   = (const int*)  d_in[4];   // [N]
    const float* W_res = (const float*)d_in[5];   // [21,256]

    float* out       = (float*)d_out;
    float* node_attr = out;                            // N*256
    float* out_dst   = node_attr + N_NODES * EMBED;    // N*10
    float* out_dist  = out_dst   + N_NODES * KNEI;     // N*10
    float* out_valid = out_dist  + N_NODES * KNEI;     // N*10

    float* D = (float*)d_ws;                           // N*N floats (~37.7 MB)

    node_attr_kernel<<<(N_NODES * EMBED) / 256, 256, 0, stream>>>(W_res, S, RP, node_attr);

    // 192 i-tiles x 48 j-groups (of 4 tiles) = 9216 waves; 8 waves/block
    dist_kernel<<<(NTI * (NTI / JT)) / 8, 256, 0, stream>>>(X, Seg, bid, D);

    topk_kernel<<<(N_NODES + 255) / 256, 256, 0, stream>>>(D, out_dst, out_dist, out_valid);
}